// HyperLayer_31868657336333
// MI455X (gfx1250) — compile-verified
//
#include <hip/hip_runtime.h>

// ---------------------------------------------------------------------------
// HyperLayer for MI455X (gfx1250, wave32).
// Density exponent computed as a K=8 (padded from 5) f32 GEMM using
// V_WMMA_F32_16X16X4_F32 (two chained K=4 WMMAs), exp via v_exp_f32.
// Duplicate mask folded into feature slot 5 with a 1e30 penalty so EXEC
// stays all-ones across WMMA. Tile loops use readfirstlane-pinned scalar
// control flow so EXEC is statically untouched around WMMA.
// ---------------------------------------------------------------------------

typedef float v2f __attribute__((ext_vector_type(2)));
typedef float v8f __attribute__((ext_vector_type(8)));

#define B_      4
#define K_      256
#define GADD    64
#define RADD    64
#define PPK     132                 // 4 + 64 + 64 points per gaussian
#define NPTS    (K_ * PPK)          // 33792 points per batch
#define NTILES  (NPTS / 16)         // 2112 row tiles
#define KTILES  (K_ / 16)           // 16 column tiles
#define OUTS    4096
#define INS     4096
#define EPS_    1e-6f
#define HASH_LG 17
#define HASH_SZ (1 << HASH_LG)      // 131072 slots per batch (load ~0.26)
#define BIGPEN  1.0e30f

__device__ __forceinline__ v8f wmma4(v2f a, v2f b, v8f c) {
  // (neg_a, A, neg_b, B, c_mod, C, reuse_a, reuse_b)
  return __builtin_amdgcn_wmma_f32_16x16x4_f32(false, a, false, b, (short)0, c,
                                               false, false);
}

// --------------------------- K0: init --------------------------------------
__global__ void hl_init(const float* __restrict__ means,
                        const float* __restrict__ sigmas,
                        const float* __restrict__ bias,
                        float* __restrict__ y,
                        float* __restrict__ gvec,
                        float* __restrict__ colsum,
                        long long* __restrict__ hkey,
                        int* __restrict__ hval) {
  int i = blockIdx.x * blockDim.x + threadIdx.x;
  if (i < B_ * HASH_SZ) { hkey[i] = -1ll; hval[i] = 0x7FFFFFFF; }
  if (i < B_ * OUTS)    { y[i] = bias[i & (OUTS - 1)]; }
  if (i < B_ * K_) {
    float m0 = means[i * 2 + 0], m1 = means[i * 2 + 1];
    float inv0 = 1.0f / (EPS_ + sigmas[i * 2 + 0]);
    float inv1 = 1.0f / (EPS_ + sigmas[i * 2 + 1]);
    float* g = gvec + (size_t)i * 8;
    g[0] = inv0;               g[1] = inv1;
    g[2] = -2.0f * m0 * inv0;  g[3] = -2.0f * m1 * inv1;
    g[4] = m0 * m0 * inv0 + m1 * m1 * inv1;
    g[5] = BIGPEN;             g[6] = 0.0f;  g[7] = 0.0f;
    colsum[i] = 0.0f;
  }
}

// --------------------- K1: generate points + hash insert -------------------
__global__ void hl_gen(const float* __restrict__ means,
                       const float* __restrict__ rr_u,
                       const float* __restrict__ g_u,
                       float* __restrict__ feat,
                       int* __restrict__ oidx,
                       int* __restrict__ iidx,
                       long long* __restrict__ hkey,
                       int* __restrict__ hval) {
  int gid = blockIdx.x * blockDim.x + threadIdx.x;
  if (gid >= B_ * NPTS) return;
  int b = gid / NPTS;
  int n = gid - b * NPTS;
  int k = n / PPK;
  int j = n - k * PPK;

  float m0 = means[(b * K_ + k) * 2 + 0];
  float m1 = means[(b * K_ + k) * 2 + 1];
  int t0, t1;
  if (j < 4) {
    // FLOOR_MASK order: (T,T),(T,F),(F,T),(F,F); True -> floor
    t0 = (int)((j < 2)        ? floorf(m0) : ceilf(m0));
    t1 = (int)(((j & 1) == 0) ? floorf(m1) : ceilf(m1));
  } else if (j < 4 + GADD) {
    const float* gu = g_u + (((size_t)(b * K_ + k)) * GADD + (j - 4)) * 2;
    t0 = (int)floorf(gu[0] * (1.0f - EPS_) * (float)OUTS);
    t1 = (int)floorf(gu[1] * (1.0f - EPS_) * (float)INS);
  } else {
    const float* ru = rr_u + (((size_t)(b * K_ + k)) * RADD + (j - 4 - GADD)) * 2;
    float mr0 = rintf(m0), mr1 = rintf(m1);       // round-half-even == jnp.round
    float lo0 = mr0 - 8.0f, up0 = mr0 + 8.0f;
    lo0 = lo0 < 0.0f ? 0.0f : lo0;
    if (up0 > (float)OUTS) lo0 = (float)OUTS - 16.0f;
    float lo1 = mr1 - 8.0f, up1 = mr1 + 8.0f;
    lo1 = lo1 < 0.0f ? 0.0f : lo1;
    if (up1 > (float)INS) lo1 = (float)INS - 16.0f;
    t0 = (int)(ru[0] * (1.0f - EPS_) * 16.0f + lo0);
    t1 = (int)(ru[1] * (1.0f - EPS_) * 16.0f + lo1);
  }
  oidx[gid] = t0;
  iidx[gid] = t1;

  float p0 = (float)t0, p1 = (float)t1;
  float* f = feat + (size_t)gid * 8;
  f[0] = p0 * p0; f[1] = p1 * p1; f[2] = p0; f[3] = p1;
  f[4] = 1.0f;    f[5] = 0.0f;    f[6] = 0.0f; f[7] = 0.0f;

  // reference duplicate key: (t0+1)^2 * (t1+1)^3  (int64, collisions intended)
  long long a = (long long)(t0 + 1);
  long long c = (long long)(t1 + 1);
  long long key = a * a * c * c * c;
  unsigned h = (unsigned)(((unsigned long long)key * 0x9E3779B97F4A7C15ull)
                          >> (64 - HASH_LG)) & (HASH_SZ - 1);
  long long* tk = hkey + (size_t)b * HASH_SZ;
  int*       tv = hval + (size_t)b * HASH_SZ;
  for (;;) {
    long long prev = (long long)atomicCAS((unsigned long long*)&tk[h],
                                          (unsigned long long)(-1ll),
                                          (unsigned long long)key);
    if (prev == -1ll || prev == key) { atomicMin(&tv[h], n); break; }
    h = (h + 1) & (HASH_SZ - 1);
  }
}

// --------------------- K2: mark duplicates into feature[5] -----------------
__global__ void hl_dup(const int* __restrict__ oidx,
                       const int* __restrict__ iidx,
                       const long long* __restrict__ hkey,
                       const int* __restrict__ hval,
                       float* __restrict__ feat) {
  int gid = blockIdx.x * blockDim.x + threadIdx.x;
  if (gid >= B_ * NPTS) return;
  int b = gid / NPTS;
  int n = gid - b * NPTS;
  long long a = (long long)(oidx[gid] + 1);
  long long c = (long long)(iidx[gid] + 1);
  long long key = a * a * c * c * c;
  unsigned h = (unsigned)(((unsigned long long)key * 0x9E3779B97F4A7C15ull)
                          >> (64 - HASH_LG)) & (HASH_SZ - 1);
  const long long* tk = hkey + (size_t)b * HASH_SZ;
  const int*       tv = hval + (size_t)b * HASH_SZ;
  for (;;) {
    if (tk[h] == key) break;
    h = (h + 1) & (HASH_SZ - 1);
  }
  feat[(size_t)gid * 8 + 5] = (tv[h] != n) ? 1.0f : 0.0f;
}

// --------------------- K3: column sums of exp tile (WMMA) ------------------
__global__ void __launch_bounds__(256)
hl_colsum(const float* __restrict__ feat,
          const float* __restrict__ gvec,
          float* __restrict__ colsum) {
  __shared__ float red[8 * 16];
  int kt   = blockIdx.x;
  int b    = blockIdx.z;
  int wave = threadIdx.x >> 5;
  int lane = threadIdx.x & 31;
  int hi   = lane >> 4;       // half-wave selects K slice of A/B fragment
  int ln   = lane & 15;       // row (A) / column (B,C) index

  const v2f* gv = (const v2f*)(gvec + ((size_t)b * K_ + kt * 16 + ln) * 8);
  v2f bf0 = gv[hi];           // lanes<16: (inv0,inv1)     lanes>=16: (-2m0i0,-2m1i1)
  v2f bf1 = gv[2 + hi];       // lanes<16: (c, 1e30)       lanes>=16: (0,0)

  // wave-uniform tile loop, pinned scalar so EXEC stays all-ones across WMMA
  int tstart = __builtin_amdgcn_readfirstlane(blockIdx.y * 8 + wave);
  int stride = gridDim.y * 8;

  float colacc = 0.0f;
  for (int t = tstart; t < NTILES; t += stride) {
    const v2f* fp = (const v2f*)(feat + ((size_t)b * NPTS + t * 16 + ln) * 8);
    v2f af0 = fp[hi];         // lanes<16: (p0^2,p1^2)     lanes>=16: (p0,p1)
    v2f af1 = fp[2 + hi];     // lanes<16: (1, dup)        lanes>=16: (0,0)
    v8f cc = {};
    cc = wmma4(af0, bf0, cc);
    cc = wmma4(af1, bf1, cc);
#pragma unroll
    for (int q = 0; q < 8; ++q) colacc += __expf(-0.5f * cc[q]);
  }
  colacc += __shfl_xor(colacc, 16);        // merge M=0..7 and M=8..15 halves
  if (lane < 16) red[wave * 16 + lane] = colacc;
  __syncthreads();
  if (threadIdx.x < 16) {
    float s = 0.0f;
#pragma unroll
    for (int w = 0; w < 8; ++w) s += red[w * 16 + threadIdx.x];
    atomicAdd(&colsum[b * K_ + kt * 16 + threadIdx.x], s);
  }
}

// --------------------- K4: vscaled = values / colsum -----------------------
__global__ void hl_vscale(const float* __restrict__ values,
                          const float* __restrict__ colsum,
                          float* __restrict__ vscaled) {
  int i = blockIdx.x * blockDim.x + threadIdx.x;
  if (i < B_ * K_) vscaled[i] = values[i] / colsum[i];
}

// --------------- K5: weights (WMMA) + gather/scatter into y ----------------
__global__ void __launch_bounds__(256)
hl_weights(const float* __restrict__ feat,
           const float* __restrict__ gvec,
           const float* __restrict__ vscaled,
           const int* __restrict__ oidx,
           const int* __restrict__ iidx,
           const float* __restrict__ x,
           float* __restrict__ y) {
  __shared__ float wrow[8 * 16];
  int b    = blockIdx.y;
  int wave = threadIdx.x >> 5;
  int lane = threadIdx.x & 31;
  int hi   = lane >> 4;
  int ln   = lane & 15;

  int tstart = __builtin_amdgcn_readfirstlane(blockIdx.x * 8 + wave);
  int stride = gridDim.x * 8;

  for (int t = tstart; t < NTILES; t += stride) {
    const v2f* fp = (const v2f*)(feat + ((size_t)b * NPTS + t * 16 + ln) * 8);
    v2f af0 = fp[hi];
    v2f af1 = fp[2 + hi];
    float wacc[8];
#pragma unroll
    for (int q = 0; q < 8; ++q) wacc[q] = 0.0f;

#pragma unroll 2
    for (int kt = 0; kt < KTILES; ++kt) {
      const v2f* gv = (const v2f*)(gvec + ((size_t)b * K_ + kt * 16 + ln) * 8);
      v2f bf0 = gv[hi];
      v2f bf1 = gv[2 + hi];
      float vs = vscaled[b * K_ + kt * 16 + ln];
      v8f cc = {};
      cc = wmma4(af0, bf0, cc);
      cc = wmma4(af1, bf1, cc);
#pragma unroll
      for (int q = 0; q < 8; ++q) wacc[q] += __expf(-0.5f * cc[q]) * vs;
    }
    // reduce over the 16 columns held across lanes of each half-wave
#pragma unroll
    for (int m = 1; m < 16; m <<= 1) {
#pragma unroll
      for (int q = 0; q < 8; ++q) wacc[q] += __shfl_xor(wacc[q], m);
    }
    if (ln == 0) {  // lane 0 holds rows 0..7, lane 16 holds rows 8..15
#pragma unroll
      for (int q = 0; q < 8; ++q) wrow[wave * 16 + hi * 8 + q] = wacc[q];
    }
    asm volatile("s_wait_dscnt 0" ::: "memory");
    if (lane < 16) {
      int row  = t * 16 + lane;
      float w  = wrow[wave * 16 + lane];
      int oi   = oidx[(size_t)b * NPTS + row];
      int ii   = iidx[(size_t)b * NPTS + row];
      atomicAdd(&y[b * OUTS + oi], w * x[b * INS + ii]);
    }
  }
}

// ---------------------------------------------------------------------------
extern "C" void kernel_launch(void* const* d_in, const int* in_sizes, int n_in,
                              void* d_out, int out_size, void* d_ws,
                              size_t ws_size, hipStream_t stream) {
  (void)in_sizes; (void)n_in; (void)out_size; (void)ws_size;
  const float* x      = (const float*)d_in[0];
  const float* means  = (const float*)d_in[1];
  const float* sigmas = (const float*)d_in[2];
  const float* values = (const float*)d_in[3];
  const float* bias   = (const float*)d_in[4];
  const float* rr_u   = (const float*)d_in[5];
  const float* g_u    = (const float*)d_in[6];
  float* y = (float*)d_out;

  // workspace layout (~11.7 MB total)
  char* ws = (char*)d_ws;
  size_t off = 0;
  auto carve = [&](size_t bytes) -> char* {
    char* p = ws + off;
    off = (off + bytes + 255) & ~(size_t)255;
    return p;
  };
  float*     feat    = (float*)    carve((size_t)B_ * NPTS * 8 * 4);
  float*     gvec    = (float*)    carve((size_t)B_ * K_ * 8 * 4);
  int*       oidx    = (int*)      carve((size_t)B_ * NPTS * 4);
  int*       iidx    = (int*)      carve((size_t)B_ * NPTS * 4);
  long long* hkey    = (long long*)carve((size_t)B_ * HASH_SZ * 8);
  int*       hval    = (int*)      carve((size_t)B_ * HASH_SZ * 4);
  float*     colsum  = (float*)    carve((size_t)B_ * K_ * 4);
  float*     vscaled = (float*)    carve((size_t)B_ * K_ * 4);

  hl_init<<<(B_ * HASH_SZ + 255) / 256, 256, 0, stream>>>(
      means, sigmas, bias, y, gvec, colsum, hkey, hval);
  hl_gen<<<(B_ * NPTS + 255) / 256, 256, 0, stream>>>(
      means, rr_u, g_u, feat, oidx, iidx, hkey, hval);
  hl_dup<<<(B_ * NPTS + 255) / 256, 256, 0, stream>>>(
      oidx, iidx, hkey, hval, feat);
  hl_colsum<<<dim3(KTILES, 11, B_), 256, 0, stream>>>(feat, gvec, colsum);
  hl_vscale<<<(B_ * K_ + 255) / 256, 256, 0, stream>>>(values, colsum, vscaled);
  hl_weights<<<dim3(66, B_), 256, 0, stream>>>(feat, gvec, vscaled, oidx, iidx,
                                               x, y);
}